// AttentiveBP_47236050321586
// MI455X (gfx1250) — compile-verified
//
#include <hip/hip_runtime.h>
#include <hip/hip_bf16.h>

// ---------------- problem constants ----------------
#define NVN   8192
#define NFN   4096
#define NMSG  12288
#define NTOT  36864          // NV + NF + 2*NM
#define EEDGE 262144
#define E2    (EEDGE + NTOT) // edges + self loops = 299008
#define GG    8192
#define KK    16
#define EMBD  61
#define DOMD  16
#define HEADS 4

typedef __attribute__((ext_vector_type(2))) float v2f;
typedef __attribute__((ext_vector_type(8))) float v8f;

// ------------- helpers -------------
__device__ __forceinline__ float sigmoidf(float x) { return 1.0f / (1.0f + __expf(-x)); }
__device__ __forceinline__ float lrelu(float x, float s) { return x > 0.0f ? x : s * x; }

// order-preserving float <-> uint encoding for atomicMax on floats
__device__ __forceinline__ unsigned fenc(float f) {
    unsigned b = __float_as_uint(f);
    return (b & 0x80000000u) ? ~b : (b | 0x80000000u);
}
__device__ __forceinline__ float fdec(unsigned e) {
    unsigned b = (e & 0x80000000u) ? (e ^ 0x80000000u) : ~e;
    return __uint_as_float(b);
}

// ---------------- zero fill ----------------
__global__ void zero_f32(float* __restrict__ p, int n) {
    int i = blockIdx.x * blockDim.x + threadIdx.x;
    if (i < n) p[i] = 0.0f;
}

// ---------------- GRU: one thread per (row, hidden unit) ----------------
__global__ void gru_kernel(const float* __restrict__ x, const float* __restrict__ h,
                           const float* __restrict__ wih, const float* __restrict__ whh,
                           const float* __restrict__ bih, const float* __restrict__ bhh,
                           float* __restrict__ hout, int rows) {
    int i = blockIdx.x * blockDim.x + threadIdx.x;
    if (i >= rows * EMBD) return;
    int row = i / EMBD;
    int k   = i - row * EMBD;
    const float* xr = x + row * DOMD;
    const float* hr = h + row * EMBD;

    float gi_r = bih[k],          gh_r = bhh[k];
    float gi_z = bih[EMBD + k],   gh_z = bhh[EMBD + k];
    float gi_n = bih[2*EMBD + k], gh_n = bhh[2*EMBD + k];
#pragma unroll
    for (int j = 0; j < DOMD; ++j) {
        float xv = xr[j];
        gi_r += xv * wih[k * DOMD + j];
        gi_z += xv * wih[(EMBD + k) * DOMD + j];
        gi_n += xv * wih[(2*EMBD + k) * DOMD + j];
    }
    for (int j = 0; j < EMBD; ++j) {
        float hv = hr[j];
        gh_r += hv * whh[k * EMBD + j];
        gh_z += hv * whh[(EMBD + k) * EMBD + j];
        gh_n += hv * whh[(2*EMBD + k) * EMBD + j];
    }
    float rg = sigmoidf(gi_r + gh_r);
    float zg = sigmoidf(gi_z + gh_z);
    float ng = tanhf(gi_n + rg * gh_n);
    hout[i] = (1.0f - zg) * ng + zg * hr[k];
}

// ---------------- assemble node features X (NTOT x 64) ----------------
__global__ void build_x(const float* __restrict__ vn_prefix, const int* __restrict__ vn_colors,
                        const float* __restrict__ emb, const float* __restrict__ fn_embed,
                        const float* __restrict__ msg_prefix,
                        const float* __restrict__ v2f_h, const float* __restrict__ f2v_h,
                        float* __restrict__ X) {
    int i = blockIdx.x * blockDim.x + threadIdx.x;
    if (i >= NTOT * 64) return;
    int node = i >> 6, c = i & 63;
    float val;
    if (node < NVN) {
        val = (c < 3) ? vn_prefix[node * 3 + c]
                      : emb[vn_colors[node] * EMBD + (c - 3)];
    } else if (node < NVN + NFN) {
        val = fn_embed[(node - NVN) * 64 + c];
    } else {
        int m = node - NVN - NFN;
        if (c < 3) val = msg_prefix[m * 3 + c];
        else {
            int cc = c - 3;
            val = (m < NMSG) ? v2f_h[m * EMBD + cc] : f2v_h[(m - NMSG) * EMBD + cc];
        }
    }
    X[i] = val;
}

// ---------------- WMMA f32 GEMM: XL = X(N,CIN) @ W(CIN,COUT) ----------------
// one wave32 per 16x16 output tile, V_WMMA_F32_16X16X4_F32 over K in steps of 4.
// W is staged in LDS (shared by all 8 waves of the block).
template <int CIN, int COUT>
__global__ void gat_gemm(const float* __restrict__ X, const float* __restrict__ W,
                         float* __restrict__ XL, int nrows) {
    __shared__ float Wl[CIN * COUT];
    for (int i = threadIdx.x; i < CIN * COUT; i += blockDim.x)
        Wl[i] = W[i];
    __syncthreads();

    int wave = (blockIdx.x * blockDim.x + threadIdx.x) >> 5;
    int lane = threadIdx.x & 31;
    const int colTiles = COUT / 16;
    int m0 = (wave / colTiles) * 16;
    int n0 = (wave % colTiles) * 16;
    if (m0 >= nrows) return;

    int r  = m0 + (lane & 15);        // A row for this lane
    int c  = n0 + (lane & 15);        // B/D column for this lane
    int kh = (lane >> 4) << 1;        // 0 for lanes 0-15, 2 for lanes 16-31

    v8f acc = {};
#pragma unroll
    for (int k0 = 0; k0 < CIN; k0 += 4) {
        v2f a, b;
        a = *(const v2f*)(X + (long long)r * CIN + k0 + kh);   // global_load_b64
        b.x = Wl[(k0 + kh) * COUT + c];                        // ds_load_b32
        b.y = Wl[(k0 + kh + 1) * COUT + c];                    // ds_load_b32
        acc = __builtin_amdgcn_wmma_f32_16x16x4_f32(
            /*neg_a=*/false, a, /*neg_b=*/false, b,
            /*c_mod=*/(short)0, acc, /*reuse_a=*/false, /*reuse_b=*/false);
    }
    int rowBase = m0 + ((lane >> 4) << 3);   // +8 for upper half lanes
#pragma unroll
    for (int v = 0; v < 8; ++v)
        XL[(long long)(rowBase + v) * COUT + c] = acc[v];
}

// ---------------- per-node attention coefficients ----------------
__global__ void gat_attn_coef(const float* __restrict__ XL,
                              const float* __restrict__ as_w, const float* __restrict__ ad_w,
                              float* __restrict__ AS, float* __restrict__ AD,
                              int n, int cout) {
    int i = blockIdx.x * blockDim.x + threadIdx.x;
    if (i >= n * HEADS) return;
    int node = i >> 2, h = i & 3;
    int ch = cout >> 2;
    const float* row = XL + (long long)node * cout + h * ch;
    float s = 0.0f, d = 0.0f;
    for (int c = 0; c < ch; ++c) {
        float v = row[c];
        s += v * as_w[h * ch + c];
        d += v * ad_w[h * ch + c];
    }
    AS[i] = s;
    AD[i] = d;
}

// ---------------- edge pass 1: scores + segment max ----------------
__global__ void edge_score_max(const int* __restrict__ ei,
                               const float* __restrict__ AS, const float* __restrict__ AD,
                               float* __restrict__ EB, unsigned* __restrict__ MAXB) {
    int idx = blockIdx.x * blockDim.x + threadIdx.x;
    if (idx >= E2) return;
    int src, dst;
    if (idx < EEDGE) { src = ei[idx]; dst = ei[EEDGE + idx]; }
    else             { src = dst = idx - EEDGE; }
#pragma unroll
    for (int h = 0; h < HEADS; ++h) {
        float e = AS[src * HEADS + h] + AD[dst * HEADS + h];
        e = lrelu(e, 0.2f);
        EB[idx * HEADS + h] = e;
        atomicMax(&MAXB[dst * HEADS + h], fenc(e));
    }
}

// ---------------- edge pass 2: exp + segment sum ----------------
__global__ void edge_exp_sum(const int* __restrict__ ei, const unsigned* __restrict__ MAXB,
                             float* __restrict__ EB, float* __restrict__ DEN) {
    int idx = blockIdx.x * blockDim.x + threadIdx.x;
    if (idx >= E2) return;
    int dst = (idx < EEDGE) ? ei[EEDGE + idx] : (idx - EEDGE);
#pragma unroll
    for (int h = 0; h < HEADS; ++h) {
        float ex = __expf(EB[idx * HEADS + h] - fdec(MAXB[dst * HEADS + h]));
        EB[idx * HEADS + h] = ex;
        atomicAdd(&DEN[dst * HEADS + h], ex);
    }
}

// ---------------- edge pass 3: weighted scatter-add, 4 channels per thread ----------------
// cout/4 (head width) is a multiple of 4, so a 4-channel group stays in one head.
__global__ void edge_aggregate(const int* __restrict__ ei,
                               const float* __restrict__ EB, const float* __restrict__ DEN,
                               const float* __restrict__ XL, float* __restrict__ AGG,
                               int cout) {
    int cq = cout >> 2;  // channel quads per edge
    long long i = (long long)blockIdx.x * blockDim.x + threadIdx.x;
    long long total = (long long)E2 * cq;
    if (i >= total) return;
    int idx = (int)(i / cq);
    int c0  = ((int)(i - (long long)idx * cq)) << 2;
    int src, dst;
    if (idx < EEDGE) { src = ei[idx]; dst = ei[EEDGE + idx]; }
    else             { src = dst = idx - EEDGE; }
    int ch = cout >> 2;          // channels per head
    int h  = c0 / ch;
    float alpha = EB[idx * HEADS + h] / DEN[dst * HEADS + h];
    const float4 xv = *(const float4*)(XL + (long long)src * cout + c0);  // global_load_b128
    float* aggp = AGG + (long long)dst * cout + c0;
    atomicAdd(aggp + 0, alpha * xv.x);
    atomicAdd(aggp + 1, alpha * xv.y);
    atomicAdd(aggp + 2, alpha * xv.z);
    atomicAdd(aggp + 3, alpha * xv.w);
}

// ---------------- finalize: concat layers (bias + leaky_relu 0.01) ----------------
__global__ void finalize_concat(const float* __restrict__ AGG, const float* __restrict__ bias,
                                float* __restrict__ Xn, int n, int cout) {
    int i = blockIdx.x * blockDim.x + threadIdx.x;
    if (i >= n * cout) return;
    int c = i % cout;
    Xn[i] = lrelu(AGG[i] + bias[c], 0.01f);
}

// ---------------- finalize layer 4: mean over heads ----------------
__global__ void finalize_mean(const float* __restrict__ AGG, const float* __restrict__ bias,
                              float* __restrict__ Xn, int n) {
    int i = blockIdx.x * blockDim.x + threadIdx.x;
    if (i >= n * 64) return;
    int node = i >> 6, c = i & 63;
    const float* a = AGG + (long long)node * 256;
    float v = 0.25f * (a[c] + a[64 + c] + a[128 + c] + a[192 + c]) + bias[c];
    Xn[i] = lrelu(v, 0.01f);
}

// ---------------- attention: fold wq/wk with ws into vectors ----------------
__global__ void compute_wqv(const float* __restrict__ wq, const float* __restrict__ wk,
                            const float* __restrict__ ws,
                            float* __restrict__ WQV, float* __restrict__ WKV) {
    int i = blockIdx.x * blockDim.x + threadIdx.x;
    if (i >= HEADS * 64) return;
    int h = i >> 6, d = i & 63;
    float q = 0.0f, k = 0.0f;
    for (int e = 0; e < 64; ++e) {
        q += wq[h * 4096 + d * 64 + e] * ws[h * 128 + e];
        k += wk[h * 4096 + d * 64 + e] * ws[h * 128 + 64 + e];
    }
    WQV[i] = q;
    WKV[i] = k;
}

__global__ void compute_qk(const float* __restrict__ X,
                           const float* __restrict__ WQV, const float* __restrict__ WKV,
                           float* __restrict__ QS, float* __restrict__ KS) {
    int i = blockIdx.x * blockDim.x + threadIdx.x;
    if (i >= NFN * HEADS) return;
    int node = i >> 2, h = i & 3;
    const float* pr = X + (long long)(NVN + node) * 64;
    float q = 0.0f, k = 0.0f;
    for (int d = 0; d < 64; ++d) {
        q += pr[d] * WQV[h * 64 + d];
        k += pr[d] * WKV[h * 64 + d];
    }
    QS[i] = q;
    KS[i] = k;
}

// ---------------- final group epilogue: one thread per (g, head) ----------------
__global__ void group_out(const int* __restrict__ full, const int* __restrict__ mask,
                          const int* __restrict__ tgt, const int* __restrict__ mult,
                          const float* __restrict__ QS, const float* __restrict__ KS,
                          const float* __restrict__ att_b, float* __restrict__ out) {
    int i = blockIdx.x * blockDim.x + threadIdx.x;
    if (i >= GG * HEADS) return;
    int g = i >> 2, h = i & 3;
    int t = tgt[g];
    float qt = QS[t * HEADS + h];
    float bb = att_b[h];

    float scores[KK];
    int nbrm[KK], istgt[KK];
    int degm1 = 0;
    float ssum = 0.0f;
#pragma unroll
    for (int k = 0; k < KK; ++k) {
        int f = full[g * KK + k];
        int m = mask[g * KK + k];
        float s = sigmoidf(qt + KS[f * HEADS + h] + bb);
        scores[k] = s;
        int it = (f == t) && m;
        int nb = m && !it;
        istgt[k] = it;
        nbrm[k]  = nb;
        degm1 += nb;
        if (nb) ssum += s;
    }
    float score_sum = ssum / fmaxf((float)degm1, 1.0f);
    float trg = sigmoidf(qt + KS[t * HEADS + h] + bb);
    float mx = fmaxf(score_sum, trg);
    float e0 = __expf(score_sum - mx);
    float e1 = __expf(trg - mx);
    float tw0 = e0 / (e0 + e1);
    float tw1 = e1 / (e0 + e1);

    float wmax = -1e30f;
#pragma unroll
    for (int k = 0; k < KK; ++k)
        if (nbrm[k]) wmax = fmaxf(wmax, scores[k]);
    float wsum = 0.0f;
    float wv[KK];
#pragma unroll
    for (int k = 0; k < KK; ++k) {
        float v = nbrm[k] ? __expf(scores[k] - wmax) : 0.0f;
        wv[k] = v;
        wsum += v;
    }
    float fm = (float)mult[g];
#pragma unroll
    for (int k = 0; k < KK; ++k) {
        float o;
        if (degm1 == 0)    o = 0.0f;
        else if (istgt[k]) o = tw1;
        else if (nbrm[k])  o = (wv[k] / wsum) * tw0 * fm;
        else               o = 0.0f;
        out[(g * KK + k) * HEADS + h] = o;
    }
}

// ---------------- host orchestration ----------------
static inline int nblk(long long n, int b) { return (int)((n + b - 1) / b); }

extern "C" void kernel_launch(void* const* d_in, const int* in_sizes, int n_in,
                              void* d_out, int out_size, void* d_ws, size_t ws_size,
                              hipStream_t stream) {
    (void)in_sizes; (void)n_in; (void)out_size; (void)ws_size;
    const int*   edge_index = (const int*)d_in[0];
    const int*   vn_colors  = (const int*)d_in[1];
    const int*   group_full = (const int*)d_in[2];
    const int*   group_mask = (const int*)d_in[3];
    const int*   group_tgt  = (const int*)d_in[4];
    const int*   group_mult = (const int*)d_in[5];
    const float* vn_prefix  = (const float*)d_in[6];
    const float* fn_embed   = (const float*)d_in[7];
    const float* v2f_msgs   = (const float*)d_in[8];
    const float* v2f_hidden = (const float*)d_in[9];
    const float* f2v_msgs   = (const float*)d_in[10];
    const float* f2v_hidden = (const float*)d_in[11];
    const float* msg_prefix = (const float*)d_in[12];
    const float* emb        = (const float*)d_in[13];
    const float* gv_wih = (const float*)d_in[14];
    const float* gv_whh = (const float*)d_in[15];
    const float* gv_bih = (const float*)d_in[16];
    const float* gv_bhh = (const float*)d_in[17];
    const float* gf_wih = (const float*)d_in[18];
    const float* gf_whh = (const float*)d_in[19];
    const float* gf_bih = (const float*)d_in[20];
    const float* gf_bhh = (const float*)d_in[21];
    const float* W1  = (const float*)d_in[22];
    const float* a1s = (const float*)d_in[23];
    const float* a1d = (const float*)d_in[24];
    const float* b1  = (const float*)d_in[25];
    const float* W2  = (const float*)d_in[26];
    const float* a2s = (const float*)d_in[27];
    const float* a2d = (const float*)d_in[28];
    const float* b2  = (const float*)d_in[29];
    const float* W3  = (const float*)d_in[30];
    const float* a3s = (const float*)d_in[31];
    const float* a3d = (const float*)d_in[32];
    const float* b3  = (const float*)d_in[33];
    const float* W4  = (const float*)d_in[34];
    const float* a4s = (const float*)d_in[35];
    const float* a4d = (const float*)d_in[36];
    const float* b4  = (const float*)d_in[37];
    const float* att_wq = (const float*)d_in[38];
    const float* att_wk = (const float*)d_in[39];
    const float* att_ws = (const float*)d_in[40];
    const float* att_b  = (const float*)d_in[41];

    // outputs: out (G*K*4) | v2f_h (NM*61) | f2v_h (NM*61)
    float* outp  = (float*)d_out;
    float* v2f_h = outp + (size_t)GG * KK * HEADS;
    float* f2v_h = v2f_h + (size_t)NMSG * EMBD;

    // workspace layout (floats)
    float* X    = (float*)d_ws;                         // NTOT*64
    float* XL   = X    + (size_t)NTOT * 64;             // NTOT*256
    float* AGG  = XL   + (size_t)NTOT * 256;            // NTOT*256
    float* AS   = AGG  + (size_t)NTOT * 256;            // NTOT*4
    float* AD   = AS   + (size_t)NTOT * 4;              // NTOT*4
    unsigned* MAXB = (unsigned*)(AD + (size_t)NTOT * 4);// NTOT*4
    float* DEN  = (float*)MAXB + (size_t)NTOT * 4;      // NTOT*4
    float* EB   = DEN  + (size_t)NTOT * 4;              // E2*4
    float* WQV  = EB   + (size_t)E2 * 4;                // 256
    float* WKV  = WQV + 256;                            // 256
    float* QS   = WKV + 256;                            // NF*4
    float* KS   = QS + (size_t)NFN * 4;                 // NF*4

    const int B = 256;

    // GRUs -> directly into output regions
    gru_kernel<<<nblk((long long)NMSG * EMBD, B), B, 0, stream>>>(
        v2f_msgs, v2f_hidden, gv_wih, gv_whh, gv_bih, gv_bhh, v2f_h, NMSG);
    gru_kernel<<<nblk((long long)NMSG * EMBD, B), B, 0, stream>>>(
        f2v_msgs, f2v_hidden, gf_wih, gf_whh, gf_bih, gf_bhh, f2v_h, NMSG);

    // assemble node features
    build_x<<<nblk((long long)NTOT * 64, B), B, 0, stream>>>(
        vn_prefix, vn_colors, emb, fn_embed, msg_prefix, v2f_h, f2v_h, X);

    // ---- 4 GAT layers ----
    struct { const float *W, *as, *ad, *b; int cin, cout; } L[4] = {
        {W1, a1s, a1d, b1, 64, 32},
        {W2, a2s, a2d, b2, 32, 32},
        {W3, a3s, a3d, b3, 32, 32},
        {W4, a4s, a4d, b4, 32, 256},
    };

    for (int l = 0; l < 4; ++l) {
        int cin = L[l].cin, cout = L[l].cout;
        long long waves = (long long)(NTOT / 16) * (cout / 16);
        int gemmBlocks = nblk(waves * 32, B);
        if (cin == 64 && cout == 32)
            gat_gemm<64, 32><<<gemmBlocks, B, 0, stream>>>(X, L[l].W, XL, NTOT);
        else if (cin == 32 && cout == 32)
            gat_gemm<32, 32><<<gemmBlocks, B, 0, stream>>>(X, L[l].W, XL, NTOT);
        else
            gat_gemm<32, 256><<<gemmBlocks, B, 0, stream>>>(X, L[l].W, XL, NTOT);

        gat_attn_coef<<<nblk((long long)NTOT * HEADS, B), B, 0, stream>>>(
            XL, L[l].as, L[l].ad, AS, AD, NTOT, cout);

        zero_f32<<<nblk((long long)NTOT * cout, B), B, 0, stream>>>(AGG, NTOT * cout);
        zero_f32<<<nblk((long long)NTOT * HEADS, B), B, 0, stream>>>(DEN, NTOT * HEADS);
        zero_f32<<<nblk((long long)NTOT * HEADS, B), B, 0, stream>>>((float*)MAXB, NTOT * HEADS);

        edge_score_max<<<nblk(E2, B), B, 0, stream>>>(edge_index, AS, AD, EB, MAXB);
        edge_exp_sum<<<nblk(E2, B), B, 0, stream>>>(edge_index, MAXB, EB, DEN);
        edge_aggregate<<<nblk((long long)E2 * (cout / 4), B), B, 0, stream>>>(
            edge_index, EB, DEN, XL, AGG, cout);

        if (l < 3)
            finalize_concat<<<nblk((long long)NTOT * cout, B), B, 0, stream>>>(
                AGG, L[l].b, X, NTOT, cout);
        else
            finalize_mean<<<nblk((long long)NTOT * 64, B), B, 0, stream>>>(
                AGG, L[l].b, X, NTOT);
    }

    // ---- attention epilogue ----
    compute_wqv<<<1, 256, 0, stream>>>(att_wq, att_wk, att_ws, WQV, WKV);
    compute_qk<<<nblk((long long)NFN * HEADS, B), B, 0, stream>>>(X, WQV, WKV, QS, KS);
    group_out<<<nblk((long long)GG * HEADS, B), B, 0, stream>>>(
        group_full, group_mask, group_tgt, group_mult, QS, KS, att_b, outp);
}